// VAE_Baseline_76974403879169
// MI455X (gfx1250) — compile-verified
//
#include <hip/hip_runtime.h>
#include <hip/hip_bf16.h>
#include <math.h>

typedef float v4f __attribute__((ext_vector_type(4)));
typedef int   v4i __attribute__((ext_vector_type(4)));

#define S_ 3
#define B_ 256
#define T_ 2048
#define D_ 32
#define L_ 64
#define P_ (T_ * D_)            /* 65536 floats per (s,b) plane */
#define THREADS 256
#define NIT (P_ / (THREADS * 4))  /* 64 stages of 1024 floats */

#define LOG_NORM 3.6862316528f  /* -0.5*ln(2*pi) - ln(0.01) */
#define INV2VAR  5000.0f        /* 1/(2*0.01^2) */

#if defined(__AMDGCN__) && __has_builtin(__builtin_amdgcn_global_load_async_to_lds_b128)
#define USE_ASYNC 1
typedef __attribute__((address_space(1))) void as1_void;
typedef __attribute__((address_space(3))) void as3_void;
typedef __attribute__((address_space(1))) v4i  as1_v4i;
typedef __attribute__((address_space(3))) v4i  as3_v4i;
// (global v4i* src, LDS v4i* dst, imm offset, imm cpol)
#define CP16(dst, src)                                                     \
  __builtin_amdgcn_global_load_async_to_lds_b128(                          \
      (as1_v4i*)(as1_void*)const_cast<void*>((const void*)(src)),          \
      (as3_v4i*)(as3_void*)(void*)(dst), 0, 0)
#if __has_builtin(__builtin_amdgcn_s_wait_asynccnt)
#define WAITA(n) __builtin_amdgcn_s_wait_asynccnt(n)
#else
#define WAITA(n) asm volatile("s_wait_asynccnt " #n ::: "memory")
#endif
#else
#define USE_ASYNC 0
#endif

// ---------------------------------------------------------------------------
// Kernel 1: per-b masked SSE over T for all (s,d); deterministic reductions.
// Writes ws[4*b + {0,1,2}] = sum_d gll[s,d],  ws[4*b+3] = sum_{s,d} mse_pd.
// 3-deep async global->LDS circular pipeline (issue 2 stages ahead).
// ---------------------------------------------------------------------------
__global__ __launch_bounds__(THREADS) void vae_mse_kernel(
    const float* __restrict__ pred, const float* __restrict__ truth,
    const int* __restrict__ mask, float* __restrict__ ws) {
  const int b   = blockIdx.x;
  const int tid = threadIdx.x;

  const float* tb = truth + (size_t)b * P_;
  const int*   mb = mask  + (size_t)b * P_;
  const float* p0 = pred + ((size_t)(0 * B_ + b)) * P_;
  const float* p1 = pred + ((size_t)(1 * B_ + b)) * P_;
  const float* p2 = pred + ((size_t)(2 * B_ + b)) * P_;

  __shared__ union {
    v4f stage[3][5][THREADS];                  // 60 KB async staging
    struct {
      float part_cnt[32][THREADS / 8];
      float part_m[3][32][THREADS / 8];
      float dvals[32][4];
    } red;
  } sm;

  float cnt[4] = {0.f, 0.f, 0.f, 0.f};
  float a0[4]  = {0.f, 0.f, 0.f, 0.f};
  float a1[4]  = {0.f, 0.f, 0.f, 0.f};
  float a2[4]  = {0.f, 0.f, 0.f, 0.f};

  const int f = tid * 4;  // flat float index in the (t,d) plane, step 1024/iter

#if USE_ASYNC
  // Prime stages 0 and 1 (10 asyncs in flight per wave).
#pragma unroll
  for (int st = 0; st < 2; ++st) {
    const int fp = f + st * THREADS * 4;
    CP16(&sm.stage[st][0][tid], tb + fp);
    CP16(&sm.stage[st][1][tid], mb + fp);
    CP16(&sm.stage[st][2][tid], p0 + fp);
    CP16(&sm.stage[st][3][tid], p1 + fp);
    CP16(&sm.stage[st][4][tid], p2 + fp);
  }

  int cb = 0;  // consume-buffer index (rotates 0,1,2)
  for (int it = 0; it < NIT; ++it) {
    if (it + 2 < NIT) {
      const int fn = f + (it + 2) * THREADS * 4;
      const int nb2 = (it + 2) % 3;
      CP16(&sm.stage[nb2][0][tid], tb + fn);
      CP16(&sm.stage[nb2][1][tid], mb + fn);
      CP16(&sm.stage[nb2][2][tid], p0 + fn);
      CP16(&sm.stage[nb2][3][tid], p1 + fn);
      CP16(&sm.stage[nb2][4][tid], p2 + fn);
      WAITA(10);  // oldest 5 (stage `it`) retired; 10 newer may remain
    } else if (it + 1 < NIT) {
      WAITA(5);   // stage `it` retired; final stage still in flight
    } else {
      WAITA(0);   // everything retired
    }
    asm volatile("" ::: "memory");

    v4f tv = sm.stage[cb][0][tid];
    v4i mv = *(const v4i*)&sm.stage[cb][1][tid];
    v4f q0 = sm.stage[cb][2][tid];
    v4f q1 = sm.stage[cb][3][tid];
    v4f q2 = sm.stage[cb][4][tid];
#pragma unroll
    for (int j = 0; j < 4; ++j) {
      float m = (float)mv[j];
      float x = tv[j];
      float d0 = x - q0[j]; a0[j] += m * d0 * d0;
      float d1 = x - q1[j]; a1[j] += m * d1 * d1;
      float d2 = x - q2[j]; a2[j] += m * d2 * d2;
      cnt[j] += m;
    }
    asm volatile("" ::: "memory");
    cb = (cb + 1 == 3) ? 0 : cb + 1;
  }
#else
  for (int it = 0; it < NIT; ++it) {
    const int ff = f + it * THREADS * 4;
    v4f tv = *(const v4f*)(tb + ff);
    v4i mv = *(const v4i*)(mb + ff);
    v4f q0 = *(const v4f*)(p0 + ff);
    v4f q1 = *(const v4f*)(p1 + ff);
    v4f q2 = *(const v4f*)(p2 + ff);
#pragma unroll
    for (int j = 0; j < 4; ++j) {
      float m = (float)mv[j];
      float x = tv[j];
      float d0 = x - q0[j]; a0[j] += m * d0 * d0;
      float d1 = x - q1[j]; a1[j] += m * d1 * d1;
      float d2 = x - q2[j]; a2[j] += m * d2 * d2;
      cnt[j] += m;
    }
  }
#endif

  __syncthreads();  // staging done everywhere; repurpose LDS union

  // Deterministic scatter: (tid, j) -> unique (d, slot)
  const int slot = tid >> 3;
#pragma unroll
  for (int j = 0; j < 4; ++j) {
    const int d = (4 * tid + j) & 31;
    sm.red.part_cnt[d][slot]  = cnt[j];
    sm.red.part_m[0][d][slot] = a0[j];
    sm.red.part_m[1][d][slot] = a1[j];
    sm.red.part_m[2][d][slot] = a2[j];
  }
  __syncthreads();

  if (tid < 32) {
    const int d = tid;
    float c = 0.f, m0 = 0.f, m1 = 0.f, m2 = 0.f;
    for (int k = 0; k < THREADS / 8; ++k) {  // fixed order -> deterministic
      c  += sm.red.part_cnt[d][k];
      m0 += sm.red.part_m[0][d][k];
      m1 += sm.red.part_m[1][d][k];
      m2 += sm.red.part_m[2][d][k];
    }
    float g0, g1, g2, ms;
    if (c > 0.f) {
      const float inv = 1.f / c;
      g0 = LOG_NORM - m0 * INV2VAR * inv;
      g1 = LOG_NORM - m1 * INV2VAR * inv;
      g2 = LOG_NORM - m2 * INV2VAR * inv;
      ms = (m0 + m1 + m2) * inv;
    } else {
      g0 = g1 = g2 = ms = 0.f;
    }
    sm.red.dvals[d][0] = g0;
    sm.red.dvals[d][1] = g1;
    sm.red.dvals[d][2] = g2;
    sm.red.dvals[d][3] = ms;
  }
  __syncthreads();

  if (tid == 0) {
    float s0 = 0.f, s1 = 0.f, s2 = 0.f, s3 = 0.f;
    for (int d = 0; d < 32; ++d) {
      s0 += sm.red.dvals[d][0];
      s1 += sm.red.dvals[d][1];
      s2 += sm.red.dvals[d][2];
      s3 += sm.red.dvals[d][3];
    }
    ws[4 * b + 0] = s0;
    ws[4 * b + 1] = s1;
    ws[4 * b + 2] = s2;
    ws[4 * b + 3] = s3;
  }
}

// ---------------------------------------------------------------------------
// Kernel 2: KL sums per sample s. ws[4*B_ + s] = sum_{b,l} kld
// ---------------------------------------------------------------------------
__global__ __launch_bounds__(256) void vae_kl_kernel(
    const float* __restrict__ mu, const float* __restrict__ stdv,
    float* __restrict__ ws) {
  const int s = blockIdx.x, tid = threadIdx.x;
  const float* mus = mu   + (size_t)s * (B_ * L_);
  const float* sds = stdv + (size_t)s * (B_ * L_);
  float acc = 0.f;
  for (int k = tid; k < B_ * L_; k += 256) {
    float sd = fabsf(sds[k]);
    float m  = mus[k];
    acc += -logf(sd) + 0.5f * (sd * sd + m * m) - 0.5f;
  }
  __shared__ float r[256];
  r[tid] = acc;
  __syncthreads();
  for (int off = 128; off > 0; off >>= 1) {
    if (tid < off) r[tid] += r[tid + off];
    __syncthreads();
  }
  if (tid == 0) ws[4 * B_ + s] = r[0];
}

// ---------------------------------------------------------------------------
// Kernel 3: finalize. out = [loss, rec0..2, mse, kl0..2]
// ---------------------------------------------------------------------------
__global__ void vae_final_kernel(const float* __restrict__ ws,
                                 float* __restrict__ out) {
  if (threadIdx.x != 0 || blockIdx.x != 0) return;
  float rs0 = 0.f, rs1 = 0.f, rs2 = 0.f, mt = 0.f;
  for (int b = 0; b < B_; ++b) {
    rs0 += ws[4 * b + 0];
    rs1 += ws[4 * b + 1];
    rs2 += ws[4 * b + 2];
    mt  += ws[4 * b + 3];
  }
  float rec[3], kl[3], obj[3];
  rec[0] = rs0 / (float)(B_ * D_);
  rec[1] = rs1 / (float)(B_ * D_);
  rec[2] = rs2 / (float)(B_ * D_);
  for (int s = 0; s < 3; ++s) {
    kl[s]  = ws[4 * B_ + s] / (float)(B_ * L_);
    obj[s] = rec[s] - kl[s];
  }
  const float mse = mt / (float)(S_ * B_ * D_);
  const float mx  = fmaxf(obj[0], fmaxf(obj[1], obj[2]));
  const float lse =
      mx + logf(expf(obj[0] - mx) + expf(obj[1] - mx) + expf(obj[2] - mx));
  float loss = -lse;
  if (loss != loss) loss = -(obj[0] + obj[1] + obj[2]) / 3.0f;
  out[0] = loss;
  out[1] = rec[0];
  out[2] = rec[1];
  out[3] = rec[2];
  out[4] = mse;
  out[5] = kl[0];
  out[6] = kl[1];
  out[7] = kl[2];
}

extern "C" void kernel_launch(void* const* d_in, const int* in_sizes, int n_in,
                              void* d_out, int out_size, void* d_ws,
                              size_t ws_size, hipStream_t stream) {
  // setup_inputs order: pred_y, truth, fp_mu, fp_std, mask
  const float* pred  = (const float*)d_in[0];
  const float* truth = (const float*)d_in[1];
  const float* mu    = (const float*)d_in[2];
  const float* stdv  = (const float*)d_in[3];
  const int*   mask  = (const int*)d_in[4];
  float* ws  = (float*)d_ws;
  float* out = (float*)d_out;

  vae_mse_kernel<<<B_, THREADS, 0, stream>>>(pred, truth, mask, ws);
  vae_kl_kernel<<<S_, 256, 0, stream>>>(mu, stdv, ws);
  vae_final_kernel<<<1, 32, 0, stream>>>(ws, out);
}